// Loss_fun_24421184045291
// MI455X (gfx1250) — compile-verified
//
#include <hip/hip_runtime.h>
#include <math.h>

// ---------------------------------------------------------------------------
// Problem constants (from reference)
// ---------------------------------------------------------------------------
#define NN      100000
#define DD      256
#define VV      3
#define PP      1024
#define NEGN    1024
#define UU      2048
#define LL      (PP + NEGN)        // 2048
#define MM      (LL * VV)          // 6144 rows for both contrastive losses
#define INV_T   5.0f               // 1 / 0.2
// Every sup row has exactly (V-1) + (1024-1)*V = 3071 positives, and its
// denominator mask reduces exactly to "all but self"; all rows are valid.
#define SUP_PCNT_INV (1.0f / 3071.0f)

typedef __attribute__((ext_vector_type(2))) float v2f;
typedef __attribute__((ext_vector_type(8))) float v8f;

// ws float layout: [0]=mask_sum [1]=main_bce_sum [2]=view_bce_sum
//                  [3]=sup_loss_sum [5]=unsup_loss_sum
// zsup = ws + 64 (6144*256 f32), zuns = zsup + 6144*256

// ---------------------------------------------------------------------------
__global__ void init_acc_kernel(float* acc) {
    if (threadIdx.x < 8) acc[threadIdx.x] = 0.0f;
}

// Gather sup rows: m = l*V + v ; l<P -> pos_idx[l] else neg_idx[l-P]
__global__ void pack_sup_kernel(const float* __restrict__ proj,
                                const int* __restrict__ pos_idx,
                                const int* __restrict__ neg_idx,
                                float* __restrict__ zs) {
    int m = blockIdx.x;           // 0..6143
    int d = threadIdx.x;          // 0..255
    int l = m / VV;
    int v = m - l * VV;
    int idx = (l < PP) ? pos_idx[l] : neg_idx[l - PP];
    zs[(size_t)m * DD + d] = proj[((size_t)v * NN + idx) * DD + d];
}

// Gather + L2-normalize unsup rows: m = u*V + v, one wave per row.
__global__ void pack_unsup_kernel(const float* __restrict__ proj,
                                  const int* __restrict__ unl,
                                  float* __restrict__ zu) {
    int lane = threadIdx.x & 31;
    int wave = threadIdx.x >> 5;
    int m = blockIdx.x * 8 + wave;          // 768 blocks * 8 waves = 6144
    int u = m / VV;
    int v = m - u * VV;
    int idx = unl[u];
    const float* src = proj + ((size_t)v * NN + idx) * DD;
    float vals[8];
    float ss = 0.0f;
#pragma unroll
    for (int t = 0; t < 8; ++t) {
        vals[t] = src[lane + t * 32];
        ss += vals[t] * vals[t];
    }
#pragma unroll
    for (int off = 1; off < 32; off <<= 1) ss += __shfl_xor(ss, off);
    float inv = 1.0f / (sqrtf(ss) + 1e-8f);
    float* dst = zu + (size_t)m * DD;
#pragma unroll
    for (int t = 0; t < 8; ++t) dst[lane + t * 32] = vals[t] * inv;
}

// ---------------------------------------------------------------------------
// Fused GEMM (fp32 WMMA) + masked softmax-style row reductions.
// One wave owns a 16-row tile (i0..i0+15); A fragments staged in LDS
// (step-major, conflict-free); j-sweep software-pipelined in K-chunks of 8
// wmma steps with double-buffered A(LDS)/B(global) fragment registers.
// ---------------------------------------------------------------------------
template <bool SUP>
__global__ __launch_bounds__(128) void contrastive_kernel(
        const float* __restrict__ z, float* __restrict__ acc) {
    __shared__ float aL[4 * 16 * DD];           // 64 KB: 16KB per wave

    const int lane = threadIdx.x & 31;
    const int wave = threadIdx.x >> 5;
    const int tile_i = blockIdx.x * 4 + wave;   // 96 blocks * 4 waves = 384
    const int i0 = tile_i * 16;
    const int hi = lane >> 4;                   // half-wave id (K sub-slice)
    const int ln = lane & 15;
    const int wbase = wave * 16 * DD;

    // Stage this wave's A fragments into LDS.
    // Step s holds, per lane: A pair (row=i0+ln, k = s*4 + hi*2 + {0,1}),
    // stored at wbase + s*64 + lane*2  -> 8B/lane, conflict-free.
    const float* pA = z + (size_t)(i0 + ln) * DD + hi * 2;
#pragma unroll 8
    for (int s = 0; s < 64; ++s)
        *(v2f*)&aL[wbase + s * 64 + lane * 2] = *(const v2f*)(pA + s * 4);
    __syncthreads();

    // Row metadata for the 8 accumulator registers of this lane.
    int irow[8], inode[8], iview[8], ilab[8];
#pragma unroll
    for (int p = 0; p < 8; ++p) {
        irow[p]  = i0 + hi * 8 + p;
        inode[p] = irow[p] / VV;
        iview[p] = irow[p] - inode[p] * VV;
        ilab[p]  = (inode[p] < PP) ? 1 : 0;
    }

    float dsum[8], psum[8];
#pragma unroll
    for (int p = 0; p < 8; ++p) { dsum[p] = 0.f; psum[p] = 0.f; }

    for (int jt = 0; jt < MM / 16; ++jt) {
        // keep per-tile loads inside the loop (no LICM -> no spills)
        asm volatile("" ::: "memory");

        const int j0 = jt * 16;
        const float* pB = z + (size_t)(j0 + ln) * DD + hi * 2;

        v8f c0 = {};
        v8f c1 = {};
        v2f ab[2][8], bb[2][8];

        // prologue: fragment chunk 0
#pragma unroll
        for (int t = 0; t < 8; ++t) {
            bb[0][t] = *(const v2f*)(pB + t * 4);
            ab[0][t] = *(const v2f*)&aL[wbase + t * 64 + lane * 2];
        }

#pragma unroll
        for (int ch = 0; ch < 8; ++ch) {
            const int cur = ch & 1, nxt = cur ^ 1;
            if (ch < 7) {                        // prefetch chunk ch+1
#pragma unroll
                for (int t = 0; t < 8; ++t) {
                    const int st = (ch + 1) * 8 + t;
                    bb[nxt][t] = *(const v2f*)(pB + st * 4);
                    ab[nxt][t] = *(const v2f*)&aL[wbase + st * 64 + lane * 2];
                }
            }
#pragma unroll
            for (int t = 0; t < 8; t += 2) {     // 2 independent acc chains
                c0 = __builtin_amdgcn_wmma_f32_16x16x4_f32(
                        false, ab[cur][t], false, bb[cur][t],
                        (short)0, c0, false, false);
                c1 = __builtin_amdgcn_wmma_f32_16x16x4_f32(
                        false, ab[cur][t + 1], false, bb[cur][t + 1],
                        (short)0, c1, false, false);
            }
        }

        // Epilogue: masked accumulation for this 16x16 tile.
        const int j = j0 + ln;
        const int jnode = j / VV;
        const int jview = j - jnode * VV;
        const int jlab  = (jnode < PP) ? 1 : 0;
#pragma unroll
        for (int p = 0; p < 8; ++p) {
            float s  = (c0[p] + c1[p]) * INV_T;
            float es = __expf(s);                  // |s| <= 5 -> safe unshifted
            bool  eye = (irow[p] == j);
            bool  sn  = (inode[p] == jnode);
            bool  pos = SUP ? (sn ? (iview[p] != jview) : (ilab[p] == jlab))
                            : (sn && (iview[p] != jview));
            if (!eye) dsum[p] += es;               // denominator == all but self
            if (pos)  psum[p] += s;
        }
    }

    // Row reduction across the 16 lanes of each half-wave.
#pragma unroll
    for (int off = 1; off <= 8; off <<= 1) {
#pragma unroll
        for (int p = 0; p < 8; ++p) {
            dsum[p] += __shfl_xor(dsum[p], off);
            psum[p] += __shfl_xor(psum[p], off);
        }
    }

    if (ln == 0) {
        float lsum = 0.0f;
#pragma unroll
        for (int p = 0; p < 8; ++p)   // max-shift cancels exactly
            lsum += logf(dsum[p] + 1e-12f)
                    - psum[p] * (SUP ? SUP_PCNT_INV : 0.5f);
        atomicAdd(&acc[SUP ? 3 : 5], lsum);
    }
}

// ---------------------------------------------------------------------------
// Masked BCE for fused logits + all 3 view logits in one pass.
// ---------------------------------------------------------------------------
__device__ __forceinline__ float bce_stable(float x, float y) {
    return fmaxf(x, 0.0f) - x * y + log1pf(__expf(-fabsf(x)));
}

__global__ void bce_kernel(const float* __restrict__ fused,
                           const float* __restrict__ viewlg,
                           const float* __restrict__ labels,
                           const unsigned char* __restrict__ mask,
                           float* __restrict__ acc) {
    float msum = 0.f, mainsum = 0.f, viewsum = 0.f;
    for (int i = blockIdx.x * blockDim.x + threadIdx.x; i < NN;
         i += gridDim.x * blockDim.x) {
        float mf = mask[i] ? 1.0f : 0.0f;
        float y = labels[i];
        msum += mf;
        mainsum += bce_stable(fused[i], y) * mf;
        float vs = 0.0f;
#pragma unroll
        for (int v = 0; v < VV; ++v)
            vs += bce_stable(viewlg[(size_t)v * NN + i], y);
        viewsum += vs * mf;
    }
#pragma unroll
    for (int off = 1; off < 32; off <<= 1) {
        msum    += __shfl_xor(msum, off);
        mainsum += __shfl_xor(mainsum, off);
        viewsum += __shfl_xor(viewsum, off);
    }
    if ((threadIdx.x & 31) == 0) {
        atomicAdd(&acc[0], msum);
        atomicAdd(&acc[1], mainsum);
        atomicAdd(&acc[2], viewsum);
    }
}

__global__ void finalize_kernel(const float* __restrict__ acc,
                                float* __restrict__ out) {
    if (threadIdx.x == 0) {
        float msum   = fmaxf(acc[0], 1.0f);
        float mainl  = acc[1] / msum;
        float viewl  = acc[2] / (3.0f * msum);
        float supl   = acc[3] / (float)MM;     // all 6144 anchors valid
        float unsupl = acc[5] / (float)MM;
        out[0] = mainl + viewl + supl + 0.2f * unsupl;
        out[1] = mainl;
        out[2] = viewl;
        out[3] = supl;
        out[4] = unsupl;
    }
}

// ---------------------------------------------------------------------------
extern "C" void kernel_launch(void* const* d_in, const int* in_sizes, int n_in,
                              void* d_out, int out_size, void* d_ws, size_t ws_size,
                              hipStream_t stream) {
    const float*         fused   = (const float*)d_in[0];
    const float*         viewlg  = (const float*)d_in[1];
    const float*         proj    = (const float*)d_in[2];
    const float*         labels  = (const float*)d_in[3];
    const unsigned char* mask    = (const unsigned char*)d_in[4]; // jnp bool = 1B
    const int*           pos_idx = (const int*)d_in[5];
    const int*           neg_idx = (const int*)d_in[6];
    const int*           unl_idx = (const int*)d_in[7];
    float* out = (float*)d_out;

    float* acc  = (float*)d_ws;
    float* zsup = acc + 64;                       // 256B-aligned
    float* zuns = zsup + (size_t)MM * DD;

    init_acc_kernel<<<1, 32, 0, stream>>>(acc);
    pack_sup_kernel<<<MM, DD, 0, stream>>>(proj, pos_idx, neg_idx, zsup);
    pack_unsup_kernel<<<MM / 8, 256, 0, stream>>>(proj, unl_idx, zuns);
    bce_kernel<<<256, 256, 0, stream>>>(fused, viewlg, labels, mask, acc);
    contrastive_kernel<true><<<MM / 16 / 4, 128, 0, stream>>>(zsup, acc);
    contrastive_kernel<false><<<MM / 16 / 4, 128, 0, stream>>>(zuns, acc);
    finalize_kernel<<<1, 32, 0, stream>>>(acc, out);
}